// Model_2619930051455
// MI455X (gfx1250) — compile-verified
//
#include <hip/hip_runtime.h>
#include <math.h>

typedef float v2f __attribute__((ext_vector_type(2)));
typedef float v8f __attribute__((ext_vector_type(8)));

#define DIM 16
#define HID 32
#define WAVES_PER_BLOCK 8
#define HSTRIDE 34   // 32 + 2 pad (even -> float2-aligned LDS reads)
#define XSTRIDE 17   // 16 + 1 pad

// One wave32 processes 16 consecutive tokens as 16x16 WMMA tiles.
//   GEMM1: H(16x32) = LN(X)(16x16) @ W1^T(16x32)   -> 2 tiles x 4 x wmma_f32_16x16x4
//   GEMM2: O(16x16) = GELU(H)(16x32) @ W2^T(32x16) -> 8 chained wmma_f32_16x16x4
// A-layout (f32 16x4): lane = M (token), VGPR v + lane-half h -> K = 4*kc + 2*h + v
// C/D-layout: lane = N, VGPR r + half h -> M = r + 8*h
__global__ __launch_bounds__(256) void mlp_ln_gelu_wmma(
    const float* __restrict__ x,
    const float* __restrict__ ln_w,
    const float* __restrict__ ln_b,
    const float* __restrict__ w1,   // [HID][DIM]
    const float* __restrict__ b1,   // [HID]
    const float* __restrict__ w2,   // [DIM][HID]
    const float* __restrict__ b2,   // [DIM]
    float* __restrict__ out,
    int numGroups)                  // tokens / 16
{
    __shared__ float ldsH[WAVES_PER_BLOCK][16 * HSTRIDE];
    __shared__ float ldsX[WAVES_PER_BLOCK][16 * XSTRIDE];

    const int lane = threadIdx.x & 31;
    const int wv   = threadIdx.x >> 5;
    const int t    = lane & 15;   // token-in-group (A rows) / N column (C/D)
    const int h    = lane >> 4;   // half-wave selector

    float* Hbuf = ldsH[wv];
    float* Xbuf = ldsX[wv];

    // ---- hoisted, loop-invariant weight loads in WMMA B-layout ----
    // GEMM1 B: B[k][n] = w1[n][k]; n = tile*16 + t (lane), k = kc*4 + 2h + {0,1}
    v2f w1b[2][4];
    #pragma unroll
    for (int tile = 0; tile < 2; ++tile)
        #pragma unroll
        for (int kc = 0; kc < 4; ++kc)
            w1b[tile][kc] = *(const v2f*)(w1 + (tile*16 + t)*DIM + kc*4 + 2*h);

    // GEMM2 B: B[k][n] = w2[n][k]; n = t, k = kc*4 + 2h + {0,1}
    v2f w2b[8];
    #pragma unroll
    for (int kc = 0; kc < 8; ++kc)
        w2b[kc] = *(const v2f*)(w2 + t*HID + kc*4 + 2*h);

    // LayerNorm affine params for this lane's K slots
    v2f lw[4], lb[4];
    #pragma unroll
    for (int kc = 0; kc < 4; ++kc) {
        lw[kc] = *(const v2f*)(ln_w + kc*4 + 2*h);
        lb[kc] = *(const v2f*)(ln_b + kc*4 + 2*h);
    }
    const float b1v0 = b1[t];          // bias depends on N only -> C-layout broadcast
    const float b1v1 = b1[16 + t];
    const float b2v  = b2[t];

    const int gw = blockIdx.x * WAVES_PER_BLOCK + wv;
    const int nw = gridDim.x * WAVES_PER_BLOCK;

    for (int g = gw; g < numGroups; g += nw) {
        const float* xg = x + (size_t)g * 16 * DIM;

        // ---- load X in A-layout, LayerNorm across the split halves ----
        v2f xa[4];
        float s = 0.f, ss = 0.f;
        #pragma unroll
        for (int kc = 0; kc < 4; ++kc) {
            xa[kc] = *(const v2f*)(xg + t*DIM + kc*4 + 2*h);
            s  += xa[kc].x + xa[kc].y;
            ss += xa[kc].x*xa[kc].x + xa[kc].y*xa[kc].y;
        }
        s  += __shfl_xor(s, 16, 32);   // combine lane l <-> l+16 partials
        ss += __shfl_xor(ss, 16, 32);
        const float mean = s * (1.f/16.f);
        const float var  = ss * (1.f/16.f) - mean*mean;
        const float rstd = rsqrtf(var + 1e-5f);

        v2f an[4];
        #pragma unroll
        for (int kc = 0; kc < 4; ++kc) {
            // stash raw x (for residual) in LDS, A-layout positions
            Xbuf[t*XSTRIDE + kc*4 + 2*h]     = xa[kc].x;
            Xbuf[t*XSTRIDE + kc*4 + 2*h + 1] = xa[kc].y;
            an[kc].x = (xa[kc].x - mean)*rstd*lw[kc].x + lb[kc].x;
            an[kc].y = (xa[kc].y - mean)*rstd*lw[kc].y + lb[kc].y;
        }

        // ---- GEMM1: two 16x16 tiles, K=16 via 4 chained f32 WMMAs each ----
        #pragma unroll
        for (int tile = 0; tile < 2; ++tile) {
            const float bi = tile ? b1v1 : b1v0;
            v8f acc = { bi, bi, bi, bi, bi, bi, bi, bi };
            #pragma unroll
            for (int kc = 0; kc < 4; ++kc)
                acc = __builtin_amdgcn_wmma_f32_16x16x4_f32(
                          false, an[kc], false, w1b[tile][kc],
                          (short)0, acc, false, false);
            // exact GELU on C-layout result, stage to LDS for the A-layout reload
            #pragma unroll
            for (int r = 0; r < 8; ++r) {
                const int M = r + 8*h;
                const float v = acc[r];
                Hbuf[M*HSTRIDE + tile*16 + t] =
                    0.5f * v * (1.f + erff(v * 0.70710678118654752f));
            }
        }

        asm volatile("s_wait_dscnt 0" ::: "memory");   // LDS store -> load ordering

        // ---- GEMM2: K=32 via 8 chained f32 WMMAs ----
        v8f acc2 = { b2v, b2v, b2v, b2v, b2v, b2v, b2v, b2v };
        #pragma unroll
        for (int kc = 0; kc < 8; ++kc) {
            v2f ga = *(const v2f*)(Hbuf + t*HSTRIDE + kc*4 + 2*h);
            acc2 = __builtin_amdgcn_wmma_f32_16x16x4_f32(
                       false, ga, false, w2b[kc], (short)0, acc2, false, false);
        }

        // ---- residual add + coalesced store (C-layout) ----
        float* og = out + (size_t)g * 16 * DIM;
        #pragma unroll
        for (int r = 0; r < 8; ++r) {
            const int M = r + 8*h;
            og[M*DIM + t] = acc2[r] + Xbuf[M*XSTRIDE + t];
        }
    }
}

extern "C" void kernel_launch(void* const* d_in, const int* in_sizes, int n_in,
                              void* d_out, int out_size, void* d_ws, size_t ws_size,
                              hipStream_t stream) {
    const float* x    = (const float*)d_in[0];
    const float* ln_w = (const float*)d_in[1];
    const float* ln_b = (const float*)d_in[2];
    const float* w1   = (const float*)d_in[3];
    const float* b1   = (const float*)d_in[4];
    const float* w2   = (const float*)d_in[5];
    const float* b2   = (const float*)d_in[6];
    float* out        = (float*)d_out;

    const int tokens    = in_sizes[0] / DIM;   // 32768 * 64 = 2,097,152
    const int numGroups = tokens / 16;         // divisible for this shape

    int blocks = 4096;                         // grid-stride: 4 groups per wave
    int maxBlocks = (numGroups + WAVES_PER_BLOCK - 1) / WAVES_PER_BLOCK;
    if (blocks > maxBlocks) blocks = maxBlocks;

    hipLaunchKernelGGL(mlp_ln_gelu_wmma, dim3(blocks), dim3(256), 0, stream,
                       x, ln_w, ln_b, w1, b1, w2, b2, out, numGroups);
}